// SimpleCGCNN_23192823398612
// MI455X (gfx1250) — compile-verified
//
#include <hip/hip_runtime.h>
#include <hip/hip_bf16.h>
#include <cmath>

#define NN 20000
#define EE 320000
#define HH 256
#define H2 512
#define LL 6
#define GGRAPH 32
#define AP 516   // LDS pitch for 512-wide tiles (516 % 64 == 4 -> conflict-free)
#define HP 260   // LDS pitch for 256-wide tiles (260 % 64 == 4 -> conflict-free)

typedef float v2f __attribute__((ext_vector_type(2)));
typedef float v8f __attribute__((ext_vector_type(8)));

__device__ __forceinline__ v8f v8zero() {
  v8f z = {0.f, 0.f, 0.f, 0.f, 0.f, 0.f, 0.f, 0.f};
  return z;
}

// V_WMMA_F32_16X16X4_F32 operand helpers (layouts per CDNA5 ISA):
//   A 16x4 : lanes 0-15 -> M=lane, K={k0,k0+1}; lanes 16-31 -> M=lane-16, K={k0+2,k0+3}
//   B 4x16 : lanes 0-15 -> K={k0,k0+1}, N=n0+lane; lanes 16-31 -> K={k0+2,k0+3}
//   C/D    : VGPR r -> M = r + 8*(lane>=16), N = n0 + (lane&15)
__device__ __forceinline__ v8f wmma4(v2f a, v2f b, v8f acc) {
  return __builtin_amdgcn_wmma_f32_16x16x4_f32(false, a, false, b, (short)0,
                                               acc, false, false);
}

__device__ __forceinline__ v2f ldsA(const float* Alds, int lda, int row, int ka) {
  v2f a;
  a[0] = Alds[row * lda + ka];
  a[1] = Alds[row * lda + ka + 1];
  return a;
}

__device__ __forceinline__ v2f ldB(const float* __restrict__ B, int ldb,
                                   int ka, int col) {
  const float* Bp = B + (size_t)ka * ldb + col;
  v2f b;
  b[0] = Bp[0];
  b[1] = Bp[ldb];
  return b;
}

__device__ __forceinline__ float softplusf(float x) {
  return (x > 20.f) ? x : log1pf(expf(x));
}

// ---------------------------------------------------------------------------
// x[n,h] = embed[atom_types[n], h]
__global__ void __launch_bounds__(256) embed_kernel(const int* __restrict__ at,
                                                    const float* __restrict__ emb,
                                                    float* __restrict__ x) {
  int i = blockIdx.x * 256 + threadIdx.x;      // N*H == 20000*256, exact grid
  int n = i >> 8, h = i & 255;
  x[i] = emb[(size_t)at[n] * HH + h];
}

// ---------------------------------------------------------------------------
// xt = x @ W + b   (per-layer).  Block = 256 thr (8 waves), 16 rows x 128 cols.
__global__ void __launch_bounds__(256) node_gemm_kernel(const float* __restrict__ x,
                                                        const float* __restrict__ W,
                                                        const float* __restrict__ bias,
                                                        float* __restrict__ xt) {
  __shared__ __align__(16) float Ald[16 * HP];
  const int tid = threadIdx.x;
  const int rowBase = blockIdx.x * 16;
  for (int q = tid; q < 16 * 64; q += 256) {   // 16 rows x 256 floats as float4
    int r = q >> 6, col = (q & 63) << 2;
    *(float4*)&Ald[r * HP + col] =
        *(const float4*)(x + (size_t)(rowBase + r) * HH + col);
  }
  __syncthreads();
  const int wave = tid >> 5, lane = tid & 31, half = lane >> 4, l16 = lane & 15;
  const int n0 = blockIdx.y * 128 + wave * 16;
  v8f acc = v8zero();
#pragma unroll 4
  for (int k0 = 0; k0 < HH; k0 += 4) {
    const int ka = k0 + (half << 1);
    acc = wmma4(ldsA(Ald, HP, l16, ka), ldB(W, HH, ka, n0 + l16), acc);
  }
  const int col = n0 + l16;
#pragma unroll
  for (int r = 0; r < 8; ++r) {
    int m = r + half * 8;
    xt[(size_t)(rowBase + m) * HH + col] = acc[r] + bias[col];
  }
}

// ---------------------------------------------------------------------------
// Edge message MLP, fully fused: 32 edges / block, 8 waves, 2 row-tiles per
// wave sharing one B fetch (halves the dominant L2 weight traffic).
// A = concat(xt[dst], xt[src]) (32x512) -> relu(A@W1+b1) -> @W2+b2 -> *ea
// -> atomic add into agg[dst].
__global__ void __launch_bounds__(256) edge_kernel(const float* __restrict__ xt,
                                                   const int* __restrict__ eidx,
                                                   const float* __restrict__ dist,
                                                   const float* __restrict__ W1,
                                                   const float* __restrict__ b1,
                                                   const float* __restrict__ W2,
                                                   const float* __restrict__ b2,
                                                   const float* __restrict__ eW,
                                                   const float* __restrict__ eB,
                                                   float* __restrict__ agg) {
  __shared__ __align__(16) float Ald[32 * AP];  // 66,048 B
  __shared__ __align__(16) float H1[32 * HP];   // 33,280 B
  __shared__ int sSrc[32], sDst[32];
  __shared__ float sD[32];
  const int tid = threadIdx.x;
  const int e0 = blockIdx.x * 32;              // E == 32*10000, exact
  if (tid < 32) {
    sSrc[tid] = eidx[e0 + tid];                // edge_index[0] = src
    sDst[tid] = eidx[EE + e0 + tid];           // edge_index[1] = dst
    sD[tid]   = dist[e0 + tid];
  }
  __syncthreads();
  // Gather A rows: first 256 cols from xt[dst], next 256 from xt[src].
  for (int q = tid; q < 32 * 128; q += 256) {  // 4096 float4 loads
    int r = q >> 7, col = (q & 127) << 2;
    int node = (col < HH) ? sDst[r] : sSrc[r];
    *(float4*)&Ald[r * AP + col] =
        *(const float4*)(xt + (size_t)node * HH + (col & (HH - 1)));
  }
  __syncthreads();
  const int wave = tid >> 5, lane = tid & 31, half = lane >> 4, l16 = lane & 15;
#pragma unroll
  for (int t = 0; t < 2; ++t) {                // GEMM1: 32x512 @ 512x256
    const int n0 = wave * 32 + t * 16;
    v8f accA = v8zero(), accB = v8zero();
#pragma unroll 4
    for (int k0 = 0; k0 < H2; k0 += 4) {
      const int ka = k0 + (half << 1);
      v2f b = ldB(W1, HH, ka, n0 + l16);       // shared by both row tiles
      accA = wmma4(ldsA(Ald, AP, l16, ka), b, accA);
      accB = wmma4(ldsA(Ald, AP, 16 + l16, ka), b, accB);
    }
    const int col = n0 + l16;
#pragma unroll
    for (int r = 0; r < 8; ++r) {
      int m = r + half * 8;
      H1[m * HP + col]        = fmaxf(accA[r] + b1[col], 0.f);
      H1[(m + 16) * HP + col] = fmaxf(accB[r] + b1[col], 0.f);
    }
  }
  __syncthreads();
#pragma unroll
  for (int t = 0; t < 2; ++t) {                // GEMM2: 32x256 @ 256x256
    const int n0 = wave * 32 + t * 16;
    v8f accA = v8zero(), accB = v8zero();
#pragma unroll 4
    for (int k0 = 0; k0 < HH; k0 += 4) {
      const int ka = k0 + (half << 1);
      v2f b = ldB(W2, HH, ka, n0 + l16);
      accA = wmma4(ldsA(H1, HP, l16, ka), b, accA);
      accB = wmma4(ldsA(H1, HP, 16 + l16, ka), b, accB);
    }
    const int col = n0 + l16;
    const float gate = eW[col];
    const float gofs = eB[col];
#pragma unroll
    for (int r = 0; r < 8; ++r) {
      int m = r + half * 8;
      float mv0 = (accA[r] + b2[col]) * (sD[m] * gate + gofs);
      float mv1 = (accB[r] + b2[col]) * (sD[m + 16] * gate + gofs);
      unsafeAtomicAdd(&agg[(size_t)sDst[m] * HH + col], mv0);
      unsafeAtomicAdd(&agg[(size_t)sDst[m + 16] * HH + col], mv1);
    }
  }
}

// ---------------------------------------------------------------------------
// Node update MLP + softplus + layernorm + residual. 32 nodes / block.
// U (post-softplus) overlays the dead A-tile region to stay within LDS budget.
__global__ void __launch_bounds__(256) update_kernel(float* __restrict__ x,
                                                     const float* __restrict__ agg,
                                                     const float* __restrict__ W1,
                                                     const float* __restrict__ b1,
                                                     const float* __restrict__ W2,
                                                     const float* __restrict__ b2,
                                                     const float* __restrict__ lng,
                                                     const float* __restrict__ lnb,
                                                     int firstLayer) {
  __shared__ __align__(16) float Ald[32 * AP];  // 66,048 B; reused for U
  __shared__ __align__(16) float H1[32 * HP];   // 33,280 B
  __shared__ float red[32][8];
  __shared__ float rowMean[32], rowRstd[32];
  float* U = Ald;                               // 32*HP floats, alias is safe:
                                                // A dead after GEMM1 barrier
  const int tid = threadIdx.x;
  const int rowBase = blockIdx.x * 32;          // N == 32*625, exact
  for (int q = tid; q < 32 * 128; q += 256) {   // A = concat(agg, x)
    int r = q >> 7, col = (q & 127) << 2;
    const float* sp = (col < HH)
        ? (agg + (size_t)(rowBase + r) * HH + col)
        : (x   + (size_t)(rowBase + r) * HH + (col - HH));
    *(float4*)&Ald[r * AP + col] = *(const float4*)sp;
  }
  __syncthreads();
  const int wave = tid >> 5, lane = tid & 31, half = lane >> 4, l16 = lane & 15;
#pragma unroll
  for (int t = 0; t < 2; ++t) {                 // GEMM1: 32x512 @ 512x256
    const int n0 = wave * 32 + t * 16;
    v8f accA = v8zero(), accB = v8zero();
#pragma unroll 4
    for (int k0 = 0; k0 < H2; k0 += 4) {
      const int ka = k0 + (half << 1);
      v2f b = ldB(W1, HH, ka, n0 + l16);
      accA = wmma4(ldsA(Ald, AP, l16, ka), b, accA);
      accB = wmma4(ldsA(Ald, AP, 16 + l16, ka), b, accB);
    }
    const int col = n0 + l16;
#pragma unroll
    for (int r = 0; r < 8; ++r) {
      int m = r + half * 8;
      H1[m * HP + col]        = fmaxf(accA[r] + b1[col], 0.f);
      H1[(m + 16) * HP + col] = fmaxf(accB[r] + b1[col], 0.f);
    }
  }
  __syncthreads();                              // last read of Ald was above
#pragma unroll
  for (int t = 0; t < 2; ++t) {                 // GEMM2: 32x256 @ 256x256
    const int n0 = wave * 32 + t * 16;
    v8f accA = v8zero(), accB = v8zero();
#pragma unroll 4
    for (int k0 = 0; k0 < HH; k0 += 4) {
      const int ka = k0 + (half << 1);
      v2f b = ldB(W2, HH, ka, n0 + l16);
      accA = wmma4(ldsA(H1, HP, l16, ka), b, accA);
      accB = wmma4(ldsA(H1, HP, 16 + l16, ka), b, accB);
    }
    const int col = n0 + l16;
#pragma unroll
    for (int r = 0; r < 8; ++r) {
      int m = r + half * 8;
      U[m * HP + col]        = softplusf(accA[r] + b2[col]);
      U[(m + 16) * HP + col] = softplusf(accB[r] + b2[col]);
    }
  }
  __syncthreads();
  // Layernorm over H=256 per row; 8 threads per row, 32 elems per thread.
  const int r = tid >> 3, seg = tid & 7;
  float s = 0.f;
#pragma unroll
  for (int j = 0; j < 32; ++j) s += U[r * HP + seg * 32 + j];
  red[r][seg] = s;
  __syncthreads();
  if (seg == 0) {
    float m = 0.f;
#pragma unroll
    for (int j = 0; j < 8; ++j) m += red[r][j];
    rowMean[r] = m * (1.f / HH);
  }
  __syncthreads();
  const float mu = rowMean[r];
  s = 0.f;
#pragma unroll
  for (int j = 0; j < 32; ++j) {
    float d = U[r * HP + seg * 32 + j] - mu;
    s += d * d;
  }
  red[r][seg] = s;
  __syncthreads();
  if (seg == 0) {
    float v = 0.f;
#pragma unroll
    for (int j = 0; j < 8; ++j) v += red[r][j];
    rowRstd[r] = rsqrtf(v * (1.f / HH) + 1e-5f);
  }
  __syncthreads();
  const float rs = rowRstd[r];
#pragma unroll
  for (int j = 0; j < 32; ++j) {
    int col = seg * 32 + j;
    float xn = (U[r * HP + col] - mu) * rs * lng[col] + lnb[col];
    size_t idx = (size_t)(rowBase + r) * HH + col;
    x[idx] = firstLayer ? xn : (x[idx] + xn);
  }
}

// ---------------------------------------------------------------------------
__global__ void __launch_bounds__(256) pool_kernel(const float* __restrict__ x,
                                                   const int* __restrict__ batch,
                                                   float* __restrict__ pooled) {
  int i = blockIdx.x * 256 + threadIdx.x;      // exact grid N*H
  int n = i >> 8, h = i & 255;
  unsafeAtomicAdd(&pooled[(size_t)batch[n] * HH + h], x[i]);
}

__global__ void __launch_bounds__(256) count_kernel(const int* __restrict__ batch,
                                                    float* __restrict__ counts) {
  int n = blockIdx.x * 256 + threadIdx.x;
  if (n < NN) unsafeAtomicAdd(&counts[batch[n]], 1.0f);
}

__global__ void __launch_bounds__(256) head_kernel(const float* __restrict__ pooled,
                                                   const float* __restrict__ counts,
                                                   const float* __restrict__ pW1,
                                                   const float* __restrict__ pb1,
                                                   const float* __restrict__ pW2,
                                                   const float* __restrict__ pb2,
                                                   const float* __restrict__ pW3,
                                                   const float* __restrict__ pb3,
                                                   float* __restrict__ out) {
  __shared__ float P[GGRAPH * HH];
  __shared__ float A1[GGRAPH * 128];
  __shared__ float A2[GGRAPH * 64];
  const int tid = threadIdx.x;
  for (int i = tid; i < GGRAPH * HH; i += 256) {
    int g = i >> 8;
    P[i] = pooled[i] / fmaxf(counts[g], 1.0f);
  }
  __syncthreads();
  for (int i = tid; i < GGRAPH * 128; i += 256) {
    int g = i >> 7, o = i & 127;
    float s = pb1[o];
    for (int k = 0; k < HH; ++k) s += P[g * HH + k] * pW1[k * 128 + o];
    A1[i] = fmaxf(s, 0.f);
  }
  __syncthreads();
  for (int i = tid; i < GGRAPH * 64; i += 256) {
    int g = i >> 6, o = i & 63;
    float s = pb2[o];
    for (int k = 0; k < 128; ++k) s += A1[g * 128 + k] * pW2[k * 64 + o];
    A2[i] = fmaxf(s, 0.f);
  }
  __syncthreads();
  if (tid < GGRAPH) {
    float s = pb3[0];
    for (int k = 0; k < 64; ++k) s += A2[tid * 64 + k] * pW3[k];
    out[tid] = softplusf(s);
  }
}

// ---------------------------------------------------------------------------
extern "C" void kernel_launch(void* const* d_in, const int* in_sizes, int n_in,
                              void* d_out, int out_size, void* d_ws, size_t ws_size,
                              hipStream_t stream) {
  (void)in_sizes; (void)n_in; (void)out_size; (void)ws_size;
  const int*   atom_types = (const int*)  d_in[0];
  const int*   edge_index = (const int*)  d_in[1];
  const float* distances  = (const float*)d_in[2];
  const int*   batch_ids  = (const int*)  d_in[3];
  const float* embed_w    = (const float*)d_in[4];
  const float* node_W = (const float*)d_in[5];
  const float* node_b = (const float*)d_in[6];
  const float* edge_W = (const float*)d_in[7];
  const float* edge_b = (const float*)d_in[8];
  const float* msg_W1 = (const float*)d_in[9];
  const float* msg_b1 = (const float*)d_in[10];
  const float* msg_W2 = (const float*)d_in[11];
  const float* msg_b2 = (const float*)d_in[12];
  const float* upd_W1 = (const float*)d_in[13];
  const float* upd_b1 = (const float*)d_in[14];
  const float* upd_W2 = (const float*)d_in[15];
  const float* upd_b2 = (const float*)d_in[16];
  const float* ln_g   = (const float*)d_in[17];
  const float* ln_b   = (const float*)d_in[18];
  const float* pW1 = (const float*)d_in[19];
  const float* pb1 = (const float*)d_in[20];
  const float* pW2 = (const float*)d_in[21];
  const float* pb2 = (const float*)d_in[22];
  const float* pW3 = (const float*)d_in[23];
  const float* pb3 = (const float*)d_in[24];
  float* out = (float*)d_out;

  float* x      = (float*)d_ws;                 // N*H
  float* xt     = x   + (size_t)NN * HH;        // N*H
  float* agg    = xt  + (size_t)NN * HH;        // N*H
  float* pooled = agg + (size_t)NN * HH;        // G*H
  float* counts = pooled + (size_t)GGRAPH * HH; // G

  embed_kernel<<<(NN * HH) / 256, 256, 0, stream>>>(atom_types, embed_w, x);

  for (int i = 0; i < LL; ++i) {
    node_gemm_kernel<<<dim3(NN / 16, 2), 256, 0, stream>>>(
        x, node_W + (size_t)i * HH * HH, node_b + (size_t)i * HH, xt);
    hipMemsetAsync(agg, 0, (size_t)NN * HH * sizeof(float), stream);
    edge_kernel<<<EE / 32, 256, 0, stream>>>(
        xt, edge_index, distances,
        msg_W1 + (size_t)i * H2 * HH, msg_b1 + (size_t)i * HH,
        msg_W2 + (size_t)i * HH * HH, msg_b2 + (size_t)i * HH,
        edge_W + (size_t)i * HH, edge_b + (size_t)i * HH, agg);
    update_kernel<<<NN / 32, 256, 0, stream>>>(
        x, agg,
        upd_W1 + (size_t)i * H2 * HH, upd_b1 + (size_t)i * HH,
        upd_W2 + (size_t)i * HH * HH, upd_b2 + (size_t)i * HH,
        ln_g + (size_t)i * HH, ln_b + (size_t)i * HH, i == 0 ? 1 : 0);
  }

  hipMemsetAsync(pooled, 0, ((size_t)GGRAPH * HH + GGRAPH) * sizeof(float), stream);
  pool_kernel<<<(NN * HH) / 256, 256, 0, stream>>>(x, batch_ids, pooled);
  count_kernel<<<(NN + 255) / 256, 256, 0, stream>>>(batch_ids, counts);
  head_kernel<<<1, 256, 0, stream>>>(pooled, counts, pW1, pb1, pW2, pb2, pW3, pb3, out);
}